// PytorchTokenFeaturer_26817775796413
// MI455X (gfx1250) — compile-verified
//
#include <hip/hip_runtime.h>
#include <stdint.h>

typedef __attribute__((ext_vector_type(2))) float v2f;
typedef __attribute__((ext_vector_type(4))) float f4;
typedef __attribute__((ext_vector_type(8))) float v8f;
typedef __attribute__((ext_vector_type(4))) unsigned int u32x4;
typedef __attribute__((ext_vector_type(4))) int i32x4;
typedef __attribute__((ext_vector_type(8))) int i32x8;

#define LSEQ 512
#define DIM 256
#define TILE_T 16
#define HALO 3
#define LDS_ROWS 24      // 22 live rows + 2 zero rows so the WMMA K=24 pad is clean
#define LDS_STRIDE 264   // 256 + 8 floats; TDM pad_enable reproduces this stride in HW

#if defined(__has_builtin)
#  if __has_builtin(__builtin_amdgcn_tensor_load_to_lds) && __has_builtin(__builtin_amdgcn_s_wait_tensorcnt)
#    define USE_TDM 1
#  endif
#endif
#ifndef USE_TDM
#  define USE_TDM 0
#endif

__global__ __launch_bounds__(256)
void token_featurer_kernel(const float* __restrict__ x, float* __restrict__ out) {
    __shared__ float lds[LDS_ROWS * LDS_STRIDE];   // 25,344 bytes

    const int bid = blockIdx.x;
    const int b   = bid >> 5;                 // 32 t-tiles per batch
    const int t0  = (bid & 31) * TILE_T;
    const int tid = threadIdx.x;

    const float* __restrict__ xb = x   + (size_t)b * LSEQ * DIM;
    float*       __restrict__ ob = out + (size_t)b * LSEQ * (3 * DIM);

    const int r0   = t0 - HALO;                             // global row of LDS row 0
    const int zlo  = (r0 < 0) ? -r0 : 0;                    // first live LDS row
    int zhi  = (LSEQ - 1) - r0; if (zhi > 21) zhi = 21;     // last live LDS row
    const int nrows = zhi - zlo + 1;

#if USE_TDM
    // ---- Zero rows the DMA will not write (K-pad rows 22..23 + edge-clamped rows) ----
    #pragma unroll
    for (int k = 0; k < 6; ++k) {
        const int j = tid + k * 256;          // 24 rows * 64 float4 slots
        const int r = j >> 6;
        if (r < zlo || r > zhi) {
            const f4 z = {0.f, 0.f, 0.f, 0.f};
            *(f4*)(&lds[r * LDS_STRIDE + (j & 63) * 4]) = z;
        }
    }

    // ---- Tensor Data Mover: DMA nrows x 256 f32 tile into LDS, HW-padded to stride 264 ----
    if (__builtin_amdgcn_readfirstlane(tid) < 32) {         // wave 0 only (scalar branch)
        const uint32_t lds_base = (uint32_t)(uintptr_t)(&lds[zlo * LDS_STRIDE]);
        const uint64_t ga = (uint64_t)(uintptr_t)(xb + (size_t)(r0 + zlo) * DIM);

        u32x4 g0;
        g0.x = 1u;                                           // count=1, user mode, no gather
        g0.y = lds_base;                                     // lds_addr (bytes)
        g0.z = (uint32_t)ga;                                 // global_addr[31:0]
        g0.w = (uint32_t)((ga >> 32) & 0x1FFFFFFu)           // global_addr[56:32]
             | (2u << 30);                                   // type = 2 ("image")

        i32x8 g1;
        g1[0] = (int)((2u << 16)                             // data_size = 4 bytes
                    | (1u << 20)                             // pad_enable
                    | (7u << 22)                             // pad_interval = 256 DWORDs
                    | (7u << 25));                           // pad_amount   = 8 DWORDs
        g1[1] = (int)((DIM & 0xFFFFu) << 16);                // tensor_dim0[15:0] @ bits 63:48
        g1[2] = (int)((DIM >> 16) | ((LSEQ & 0xFFFFu) << 16)); // tensor_dim0 hi | tensor_dim1 lo
        g1[3] = (int)((LSEQ >> 16) | ((uint32_t)DIM << 16)); // tensor_dim1 hi | tile_dim0 = 256
        g1[4] = nrows;                                       // tile_dim1 = live rows; tile_dim2 = 0
        g1[5] = DIM;                                         // tensor_dim0_stride = 256 elements
        g1[6] = 0;
        g1[7] = 0;

        const i32x4 gz = {0, 0, 0, 0};
#if __clang_major__ >= 23
        const i32x8 gz8 = {0, 0, 0, 0, 0, 0, 0, 0};
        __builtin_amdgcn_tensor_load_to_lds(g0, g1, gz, gz, gz8, 0);
#else
        __builtin_amdgcn_tensor_load_to_lds(g0, g1, gz, gz, 0);
#endif
        __builtin_amdgcn_s_wait_tensorcnt(0);                // data resident before barrier
    }
    __syncthreads();
#else
    // ---- Fallback staging: float4 loads, OOB/K-pad rows -> 0 ----
    #pragma unroll
    for (int k = 0; k < 6; ++k) {
        const int j  = tid + k * 256;
        const int r  = j >> 6;
        const int c4 = j & 63;
        const int g  = r0 + r;
        f4 v = {0.f, 0.f, 0.f, 0.f};
        if (r >= zlo && r <= zhi && g >= 0 && g < LSEQ) {
            v = *(const f4*)(xb + g * DIM + c4 * 4);
        }
        *(f4*)(&lds[r * LDS_STRIDE + c4 * 4]) = v;
    }
    __syncthreads();
#endif

    // ---- Pass-through copy out[..., 0:256] = x (from LDS, float4 stores) ----
    #pragma unroll
    for (int k = 0; k < 4; ++k) {
        const int j  = tid + k * 256;         // 16 rows * 64 float4
        const int m  = j >> 6;
        const int c4 = j & 63;
        const f4 v = *(const f4*)(&lds[(m + HALO) * LDS_STRIDE + c4 * 4]);
        *(f4*)(ob + (t0 + m) * (3 * DIM) + c4 * 4) = v;
    }

    // ---- Windowed max: thread tid owns column tid across all 16 rows ----
    {
        const int c = tid;
        if (t0 != 0 && t0 != LSEQ - TILE_T) {
            // interior t-tile (30 of 32): window always in range, unconditional 7-tap
            #pragma unroll
            for (int m = 0; m < TILE_T; ++m) {
                float mx = lds[m * LDS_STRIDE + c];
                #pragma unroll
                for (int r = 1; r < 7; ++r) {
                    mx = fmaxf(mx, lds[(m + r) * LDS_STRIDE + c]);
                }
                ob[(t0 + m) * (3 * DIM) + 2 * DIM + c] = mx;
            }
        } else {
            // edge t-tile: skip zero-filled OOB rows
            #pragma unroll
            for (int m = 0; m < TILE_T; ++m) {
                const int t = t0 + m;
                float mx = -__builtin_inff();
                #pragma unroll
                for (int off = -HALO; off <= HALO; ++off) {
                    const int g = t + off;
                    if (g >= 0 && g < LSEQ) {
                        mx = fmaxf(mx, lds[(m + off + HALO) * LDS_STRIDE + c]);
                    }
                }
                ob[t * (3 * DIM) + 2 * DIM + c] = mx;
            }
        }
    }

    // ---- Windowed sum via V_WMMA_F32_16X16X4_F32: D = Band(16x24) x X(24x16) ----
    // Band[m,kg] = 1 iff m <= kg <= m+6 (kg = LDS row = global row t0-3+kg).
    // OOB rows are zero in LDS so the band needs no edge logic; counts fix the mean.
    const int wave  = tid >> 5;
    const int lane  = tid & 31;
    const int nlo   = lane & 15;              // A-row / B-col / D-col index
    const int khalf = (lane >> 4) << 1;       // 0 for lanes 0-15, 2 for lanes 16-31

    #pragma unroll
    for (int ct = 0; ct < 2; ++ct) {
        const int c0 = (wave * 2 + ct) * 16;  // 8 waves x 2 = 16 column tiles
        v8f acc = {};
        #pragma unroll
        for (int s = 0; s < 6; ++s) {
            const int kg0 = 4 * s + khalf;
            const int kg1 = kg0 + 1;
            v2f a;
            a.x = (kg0 >= nlo && kg0 <= nlo + 6) ? 1.0f : 0.0f;
            a.y = (kg1 >= nlo && kg1 <= nlo + 6) ? 1.0f : 0.0f;
            v2f bf;
            bf.x = lds[kg0 * LDS_STRIDE + c0 + nlo];
            bf.y = lds[kg1 * LDS_STRIDE + c0 + nlo];
            acc = __builtin_amdgcn_wmma_f32_16x16x4_f32(
                /*neg_a=*/false, a, /*neg_b=*/false, bf,
                /*c_mod=*/(short)0, acc, /*reuse_a=*/false, /*reuse_b=*/false);
        }
        // D layout: VGPR i, lane holds M = i + 8*(lane>=16), N = nlo
        const int mbase = (lane >> 4) * 8;
        #pragma unroll
        for (int i = 0; i < 8; ++i) {
            const int t  = t0 + mbase + i;
            const int cl = (t < HALO) ? t : HALO;
            const int cr = ((LSEQ - 1 - t) < HALO) ? (LSEQ - 1 - t) : HALO;
            const float cnt = (float)(cl + cr + 1);
            ob[t * (3 * DIM) + DIM + c0 + nlo] = acc[i] / cnt;
        }
    }
}

extern "C" void kernel_launch(void* const* d_in, const int* in_sizes, int n_in,
                              void* d_out, int out_size, void* d_ws, size_t ws_size,
                              hipStream_t stream) {
    (void)in_sizes; (void)n_in; (void)out_size; (void)d_ws; (void)ws_size;
    const float* x = (const float*)d_in[0];
    float* out = (float*)d_out;
    dim3 grid(64 * 32);   // one block per (batch, 16-row tile)
    dim3 block(256);      // 8 wave32s
    hipLaunchKernelGGL(token_featurer_kernel, grid, block, 0, stream, x, out);
}